// Quantize_68367289418001
// MI455X (gfx1250) — compile-verified
//
#include <hip/hip_runtime.h>

#define DECAY 0.99f
#define OMD   0.01f      /* 1 - DECAY */
#define EPSV  1e-5f
#define DIM   64
#define NCODE 512
#define EPAD  65         /* padded code row: bank = (j + c) % 64, conflict-free */

typedef float v2f __attribute__((ext_vector_type(2)));
typedef float v4f __attribute__((ext_vector_type(4)));
typedef float v8f __attribute__((ext_vector_type(8)));

// workspace layout (floats): [0]=diff accum, [64..575]=counts, [576..576+32768)=embed_sum
#define WS_DIFF 0
#define WS_CNT  64
#define WS_ESUM 576
#define WS_TOTAL (WS_ESUM + DIM * NCODE)

// LDS layout (floats): ET[512*65] | Xp[8*32*65] | Nrm[512] | Idx[256 ints]
#define L_ET   0
#define L_XP   (NCODE * EPAD)                 /* 33280 */
#define L_NRM  (L_XP + 8 * 32 * EPAD)         /* 49920 */
#define L_IDX  (L_NRM + NCODE)                /* 50432 */
#define L_TOTAL (L_IDX + 256)                 /* 50688 floats = 202,752 B */

__global__ void vq_zero_ws(float* __restrict__ ws, int n) {
  int i = blockIdx.x * blockDim.x + threadIdx.x;
  if (i < n) ws[i] = 0.0f;
}

__global__ __launch_bounds__(256)
void vq_main(const float* __restrict__ input, const float* __restrict__ embed,
             float* __restrict__ q_out, float* __restrict__ ind_out,
             float* __restrict__ ws, int nmacro) {
  extern __shared__ float smem[];
  float* ET  = smem + L_ET;    // transposed codebook [512][65]: ET[j*65+c] = embed[c][j]
  float* Xp  = smem + L_XP;    // 8 waves x [32][65]
  float* Nrm = smem + L_NRM;   // ||e_j||^2
  int*   Idx = (int*)(smem + L_IDX);

  const int tid  = threadIdx.x;
  const int lane = tid & 31;
  const int wave = tid >> 5;
  const int mtile = blockIdx.x * 8 + wave;   // 32-row macro tile per wave
  const bool valid = mtile < nmacro;
  const int xbase = wave * (32 * EPAD);

  // ---- stage transposed codebook (coalesced float4 global reads) ----
  {
    const v4f* e4 = (const v4f*)embed;
#pragma unroll
    for (int i = 0; i < 32; ++i) {
      int g4 = tid + i * 256;                // 8192 float4s = 64x512
      v4f v = e4[g4];
      int g = g4 << 2;
      int c = g >> 9;                        // dim index
      int j = g & 511;                       // code index (4 consecutive)
      ET[(j + 0) * EPAD + c] = v.x;
      ET[(j + 1) * EPAD + c] = v.y;
      ET[(j + 2) * EPAD + c] = v.z;
      ET[(j + 3) * EPAD + c] = v.w;
    }
  }
  // ---- stage this wave's 32x64 X tile into padded LDS ----
  if (valid) {
    const v4f* x4 = (const v4f*)(input + (size_t)mtile * 32 * DIM);
#pragma unroll
    for (int i = 0; i < 16; ++i) {
      int f4 = lane + i * 32;                // 512 float4s = 32x64
      v4f v = x4[f4];
      int f = f4 << 2;
      int r = f >> 6;
      int c = f & 63;
      float* dst = &Xp[xbase + r * EPAD + c];
      dst[0] = v.x; dst[1] = v.y; dst[2] = v.z; dst[3] = v.w;
    }
  }
  __syncthreads();

  // ---- code norms ||e_j||^2 (conflict-free on padded rows) ----
#pragma unroll
  for (int s = 0; s < 2; ++s) {
    int j = tid + s * 256;
    float acc = 0.0f;
#pragma unroll
    for (int c = 0; c < DIM; ++c) { float e = ET[j * EPAD + c]; acc += e * e; }
    Nrm[j] = acc;
  }
  __syncthreads();

  float ddiff = 0.0f;
  if (valid) {
    // A operands for both 16-row sub-tiles (16x4 f32 WMMA layout)
    const int m = lane & 15;
    const int khalf = (lane >> 4) << 1;      // lanes>=16 carry K+2,K+3
    v2f a0[16], a1[16];
#pragma unroll
    for (int k = 0; k < 16; ++k) {
      int kk = (k << 2) + khalf;
      a0[k].x = Xp[xbase + m * EPAD + kk];
      a0[k].y = Xp[xbase + m * EPAD + kk + 1];
      a1[k].x = Xp[xbase + (m + 16) * EPAD + kk];
      a1[k].y = Xp[xbase + (m + 16) * EPAD + kk + 1];
    }

    float best0[8], best1[8];
    int   bidx0[8], bidx1[8];
#pragma unroll
    for (int r = 0; r < 8; ++r) {
      best0[r] = 3.4e38f; bidx0[r] = 0;
      best1[r] = 3.4e38f; bidx1[r] = 0;
    }

    // 32 code tiles; each B fetch feeds two independent WMMA chains
    for (int t = 0; t < 32; ++t) {
      const int col = (t << 4) + m;
      const float nrmv = Nrm[col];
      v8f c0 = {}, c1 = {};
#pragma unroll
      for (int k = 0; k < 16; ++k) {         // K = 64 via 16 fp32 WMMAs
        int kk = (k << 2) + khalf;
        v2f b;
        b.x = ET[col * EPAD + kk];           // consecutive -> single LDS access
        b.y = ET[col * EPAD + kk + 1];
        c0 = __builtin_amdgcn_wmma_f32_16x16x4_f32(
            false, a0[k], false, b, (short)0, c0, false, false);
        c1 = __builtin_amdgcn_wmma_f32_16x16x4_f32(
            false, a1[k], false, b, (short)0, c1, false, false);
      }
#pragma unroll
      for (int r = 0; r < 8; ++r) {          // score = ||e||^2 - 2 x.e
        float s0 = nrmv - 2.0f * c0[r];
        if (s0 < best0[r]) { best0[r] = s0; bidx0[r] = col; }
        float s1 = nrmv - 2.0f * c1[r];
        if (s1 < best1[r]) { best1[r] = s1; bidx1[r] = col; }
      }
    }

    // min-reduce (val,idx) across 16-lane halves; first-index tie-break
#pragma unroll
    for (int r = 0; r < 8; ++r) {
      float v0 = best0[r]; int j0 = bidx0[r];
      float v1 = best1[r]; int j1 = bidx1[r];
#pragma unroll
      for (int msk = 1; msk < 16; msk <<= 1) {
        float ov0 = __shfl_xor(v0, msk, 32); int oj0 = __shfl_xor(j0, msk, 32);
        if (ov0 < v0 || (ov0 == v0 && oj0 < j0)) { v0 = ov0; j0 = oj0; }
        float ov1 = __shfl_xor(v1, msk, 32); int oj1 = __shfl_xor(j1, msk, 32);
        if (ov1 < v1 || (ov1 == v1 && oj1 < j1)) { v1 = ov1; j1 = oj1; }
      }
      best0[r] = v0; bidx0[r] = j0;
      best1[r] = v1; bidx1[r] = j1;
    }

    // writer lanes: (lane&15) < 8 handle rows row and row+16 of the macro tile
    if ((lane & 15) < 8) {
      const int r = lane & 15;
      const int row = r + ((lane >> 4) << 3);
      const int base = mtile * 32;
#pragma unroll
      for (int half = 0; half < 2; ++half) {
        const int rr = row + half * 16;
        const int j = half ? bidx1[r] : bidx0[r];
        const float bv = half ? best1[r] : best0[r];
        ind_out[base + rr] = (float)j;
        atomicAdd(&ws[WS_CNT + j], 1.0f);
        Idx[wave * 32 + rr] = j;
        float xn = 0.0f;                      // conflict-free: distinct rows
#pragma unroll
        for (int c = 0; c < DIM; ++c) {
          float x = Xp[xbase + rr * EPAD + c];
          xn += x * x;
        }
        ddiff += xn + bv;                     // = ||x - e*||^2
      }
    }
  }
  // one diff atomic per wave
#pragma unroll
  for (int msk = 1; msk < 32; msk <<= 1) ddiff += __shfl_xor(ddiff, msk, 32);
  if (valid && lane == 0) atomicAdd(&ws[WS_DIFF], ddiff);

  __syncthreads();

  // cooperative: coalesced quantize stores + embed_sum scatter, conflict-free gathers
  if (valid) {
#pragma unroll 4
    for (int r = 0; r < 32; ++r) {
      const int j = Idx[wave * 32 + r];
      const size_t grow = (size_t)mtile * 32 + r;
      q_out[grow * DIM + lane]      = ET[j * EPAD + lane];
      q_out[grow * DIM + 32 + lane] = ET[j * EPAD + 32 + lane];
      atomicAdd(&ws[WS_ESUM + lane * NCODE + j],        Xp[xbase + r * EPAD + lane]);
      atomicAdd(&ws[WS_ESUM + (lane + 32) * NCODE + j], Xp[xbase + r * EPAD + 32 + lane]);
    }
  }
}

__global__ __launch_bounds__(512)
void vq_finalize(const float* __restrict__ cluster_size,
                 const float* __restrict__ embed_avg,
                 const float* __restrict__ ws,
                 float* __restrict__ out_diff,
                 float* __restrict__ out_new_embed,
                 float* __restrict__ out_ncs,
                 float* __restrict__ out_nea,
                 float inv_count) {
  __shared__ float red[NCODE];
  const int j = threadIdx.x;
  float ncs = cluster_size[j] * DECAY + OMD * ws[WS_CNT + j];
  out_ncs[j] = ncs;
  red[j] = ncs;
  __syncthreads();
  for (int s = 256; s > 0; s >>= 1) {
    if (j < s) red[j] += red[j + s];
    __syncthreads();
  }
  const float n = red[0];
  const float cs = (ncs + EPSV) / (n + NCODE * EPSV) * n;
  const float inv_cs = 1.0f / cs;
#pragma unroll
  for (int c = 0; c < DIM; ++c) {
    float nea = embed_avg[c * NCODE + j] * DECAY + OMD * ws[WS_ESUM + c * NCODE + j];
    out_nea[c * NCODE + j] = nea;
    out_new_embed[c * NCODE + j] = nea * inv_cs;
  }
  if (j == 0) out_diff[0] = ws[WS_DIFF] * inv_count;
}

extern "C" void kernel_launch(void* const* d_in, const int* in_sizes, int n_in,
                              void* d_out, int out_size, void* d_ws, size_t ws_size,
                              hipStream_t stream) {
  const float* input     = (const float*)d_in[0];
  const float* embed     = (const float*)d_in[1];
  const float* cluster   = (const float*)d_in[2];
  const float* embed_avg = (const float*)d_in[3];
  float* out = (float*)d_out;
  float* ws  = (float*)d_ws;

  const int total  = in_sizes[0];      // 8388608 elements
  const int N      = total / DIM;      // 131072 rows
  const int nmacro = N / 32;           // 4096 macro tiles (32 rows each)

  // output tuple layout: quantize | diff | embed_ind | new_embed | new_cluster_size | new_embed_avg
  float* q_out    = out;
  float* diff_out = out + total;
  float* ind_out  = diff_out + 1;
  float* ne_out   = ind_out + N;
  float* ncs_out  = ne_out + DIM * NCODE;
  float* nea_out  = ncs_out + NCODE;

  vq_zero_ws<<<(WS_TOTAL + 255) / 256, 256, 0, stream>>>(ws, WS_TOTAL);

  const int blocks = (nmacro + 7) / 8;                    // 8 waves/block, 1 macro tile/wave
  const size_t smem = (size_t)L_TOTAL * sizeof(float);    // 202,752 B dynamic LDS (CDNA5: 320KB/WGP)
  vq_main<<<blocks, 256, smem, stream>>>(input, embed, q_out, ind_out, ws, nmacro);

  vq_finalize<<<1, 512, 0, stream>>>(cluster, embed_avg, ws, diff_out,
                                     ne_out, ncs_out, nea_out,
                                     1.0f / (float)total);
}